// BahdanauAttention_4269197492750
// MI455X (gfx1250) — compile-verified
//
#include <hip/hip_runtime.h>
#include <math.h>

// ---------------------------------------------------------------------------
// Bahdanau additive attention for MI455X (gfx1250, wave32)
//   B=8, Tq=32, Tk=2048, Q=K=U=256
//   outputs: context [B,Tq,Q] (65536 f32) then attn [B,Tq,Tk] (524288 f32)
// ---------------------------------------------------------------------------

typedef float v2f __attribute__((ext_vector_type(2)));
typedef float v8f __attribute__((ext_vector_type(8)));

#ifndef __has_builtin
#define __has_builtin(x) 0
#endif

__device__ __forceinline__ float fast_tanh(float x) {
#if __has_builtin(__builtin_amdgcn_tanhf)
    return __builtin_amdgcn_tanhf(x);
#else
    // Native gfx1250 TRANS op; v_nop covers the TRANS result-use hazard the
    // compiler cannot see through opaque asm.
    float r;
    asm volatile("v_tanh_f32 %0, %1\n\tv_nop" : "=v"(r) : "v"(x));
    return r;
#endif
}

// ---------------------------------------------------------------------------
// fp32 WMMA GEMM:  C[M,N] = A[M,K] * B   (per blockIdx.y batch)
//   BT=true : B given as W[N,K] row-major (use W^T), ldb = K
//   BT=false: B row-major [K,N], ldb = N
// lda = K, ldc = N. One wave computes a 16x64 strip (4 N-tiles), reusing the
// A fragment across 4 V_WMMA_F32_16X16X4_F32 issues per K-step.
// Requires N % 64 == 0, M % 16 == 0, K % 4 == 0 (true for all uses here).
// ---------------------------------------------------------------------------
template <bool BT>
__global__ void wmma_gemm_f32(const float* __restrict__ A,
                              const float* __restrict__ B,
                              float* __restrict__ C,
                              int M, int N, int K,
                              long long strideA, long long strideB, long long strideC)
{
    constexpr int NT = 4;                      // 16x16 tiles per wave along N

    const int lane = threadIdx.x & 31;
    const int wave = threadIdx.x >> 5;
    const int wavesPerBlock = blockDim.x >> 5;

    const int groupsN = N >> 6;                // strips of 64 columns
    const int totalStrips = (M >> 4) * groupsN;
    const int strip = blockIdx.x * wavesPerBlock + wave;
    if (strip >= totalStrips) return;          // wave-uniform -> EXEC all-1s

    const int tm = strip / groupsN;
    const int tg = strip - tm * groupsN;
    const int row0 = tm << 4;
    const int col0 = tg << 6;

    const float* __restrict__ Ab = A + (long long)blockIdx.y * strideA;
    const float* __restrict__ Bb = B + (long long)blockIdx.y * strideB;
    float* __restrict__ Cb       = C + (long long)blockIdx.y * strideC;

    const int m     = lane & 15;        // row (A) / col (B) within tile
    const int khalf = (lane >> 4) << 1; // 0 for lanes 0-15, 2 for lanes 16-31

    v8f c[NT] = {};

    #pragma unroll 4
    for (int k = 0; k < K; k += 4) {
        const int ka = k + khalf;
        v2f a;
        // A 16x4 fragment: lane m holds (K=ka, K=ka+1) of row row0+m
        a.x = Ab[(long long)(row0 + m) * K + ka];
        a.y = Ab[(long long)(row0 + m) * K + ka + 1];

        #pragma unroll
        for (int t = 0; t < NT; ++t) {
            const int cn = col0 + (t << 4) + m;
            v2f b;
            if (BT) {
                // B[k][n] = W[n*K + k]
                b.x = Bb[(long long)cn * K + ka];
                b.y = Bb[(long long)cn * K + ka + 1];
            } else {
                b.x = Bb[(long long)(ka)     * N + cn];
                b.y = Bb[(long long)(ka + 1) * N + cn];
            }
            c[t] = __builtin_amdgcn_wmma_f32_16x16x4_f32(false, a, false, b,
                                                         (short)0, c[t],
                                                         false, false);
        }
    }

    // C/D layout: VGPR j, lanes 0-15 -> M=j, lanes 16-31 -> M=j+8, N=lane&15
    const int rowOff = (lane >> 4) << 3;
    #pragma unroll
    for (int t = 0; t < NT; ++t) {
        #pragma unroll
        for (int j = 0; j < 8; ++j) {
            Cb[(long long)(row0 + rowOff + j) * N + col0 + (t << 4) + (lane & 15)]
                = c[t][j];
        }
    }
}

// ---------------------------------------------------------------------------
// scores[b,q,k] = sum_u tanh(q_proj[b,q,u] + k_proj[b,k,u]) * v[u]
// lane == q (Tq==32==wave32). Each wave handles 2 k-values; block = 8 waves.
// grid = (Tk/16, B). LDS: qp transposed (conflict-free), kp + v broadcast.
// ---------------------------------------------------------------------------
__global__ void score_tanh_kernel(const float* __restrict__ qp,   // [B,32,256]
                                  const float* __restrict__ kp,   // [B,2048,256]
                                  const float* __restrict__ v,    // [256]
                                  float* __restrict__ sc)         // [B,32,2048]
{
    constexpr int U  = 256;
    constexpr int TQ = 32;
    constexpr int TK = 2048;
    constexpr int KW = 2;      // k-values per wave

    __shared__ float qps[U * TQ];        // transposed: qps[u*32 + q]  (32 KB)
    __shared__ float vs[U];              // 1 KB
    __shared__ float kps[8 * KW * U];    // 16 KB

    const int tid  = threadIdx.x;        // 0..255
    const int lane = tid & 31;           // == q
    const int wave = tid >> 5;           // 0..7
    const int b    = blockIdx.y;
    const int k0   = blockIdx.x * (8 * KW) + wave * KW;

    // stage q_proj[b] transposed into LDS (coalesced global reads)
    const float* qpb = qp + (long long)b * TQ * U;
    for (int i = tid; i < TQ * U; i += 256) {
        const int q = i >> 8;            // i / 256
        const int u = i & 255;
        qps[u * TQ + q] = qpb[i];
    }
    vs[tid] = v[tid];

    // stage this wave's kp rows (coalesced)
    #pragma unroll
    for (int j = 0; j < KW; ++j) {
        const float* kpr = kp + ((long long)b * TK + (k0 + j)) * U;
        for (int i = lane; i < U; i += 32)
            kps[(wave * KW + j) * U + i] = kpr[i];
    }
    __syncthreads();

    float acc0 = 0.f, acc1 = 0.f;
    const int kb0 = (wave * KW + 0) * U;
    const int kb1 = (wave * KW + 1) * U;

    #pragma unroll 4
    for (int u = 0; u < U; ++u) {
        const float qv = qps[u * TQ + lane];   // per-lane, conflict-free
        const float vv = vs[u];                // uniform -> broadcast
        acc0 += fast_tanh(qv + kps[kb0 + u]) * vv;
        acc1 += fast_tanh(qv + kps[kb1 + u]) * vv;
    }

    float* srow = sc + ((long long)b * TQ + lane) * TK + k0;
    srow[0] = acc0;
    srow[1] = acc1;
}

// ---------------------------------------------------------------------------
// Row softmax over Tk=2048. One 256-thread block per (b,q) row (256 rows).
// ---------------------------------------------------------------------------
__global__ void softmax_rows_kernel(const float* __restrict__ sc,
                                    float* __restrict__ attn)
{
    constexpr int TK = 2048;
    const int row = blockIdx.x;
    const int tid = threadIdx.x;
    const float* s = sc   + (long long)row * TK;
    float*       a = attn + (long long)row * TK;

    __shared__ float red[256];

    float m = -INFINITY;
    #pragma unroll
    for (int j = 0; j < 8; ++j) m = fmaxf(m, s[tid + j * 256]);
    red[tid] = m;
    __syncthreads();
    for (int off = 128; off > 0; off >>= 1) {
        if (tid < off) red[tid] = fmaxf(red[tid], red[tid + off]);
        __syncthreads();
    }
    const float mx = red[0];
    __syncthreads();

    float e[8];
    float sum = 0.f;
    #pragma unroll
    for (int j = 0; j < 8; ++j) {
        e[j] = __expf(s[tid + j * 256] - mx);
        sum += e[j];
    }
    red[tid] = sum;
    __syncthreads();
    for (int off = 128; off > 0; off >>= 1) {
        if (tid < off) red[tid] += red[tid + off];
        __syncthreads();
    }
    const float inv = 1.0f / red[0];
    #pragma unroll
    for (int j = 0; j < 8; ++j) a[tid + j * 256] = e[j] * inv;
}

// ---------------------------------------------------------------------------
extern "C" void kernel_launch(void* const* d_in, const int* in_sizes, int n_in,
                              void* d_out, int out_size, void* d_ws, size_t ws_size,
                              hipStream_t stream)
{
    (void)in_sizes; (void)n_in; (void)out_size; (void)ws_size;

    const float* query = (const float*)d_in[0]; // [8,32,256]
    const float* keys  = (const float*)d_in[1]; // [8,2048,256]
    const float* Wq    = (const float*)d_in[2]; // [256,256]
    const float* Wk    = (const float*)d_in[3]; // [256,256]
    const float* v     = (const float*)d_in[4]; // [256]

    float* context = (float*)d_out;            // 8*32*256   = 65536
    float* attn    = (float*)d_out + 65536;    // 8*32*2048  = 524288

    float* qp = (float*)d_ws;                  // 65536 f32
    float* kp = qp + 65536;                    // 4194304 f32 (16 MB)
    float* sc = kp + (size_t)16384 * 256;      // 524288 f32 (2 MB)

    // 1) q_proj = query * Wq^T   (M=256, N=256, K=256)
    //    strips = 16 * 4 = 64, 4 waves/block -> 16 blocks
    wmma_gemm_f32<true><<<dim3(16, 1), 128, 0, stream>>>(
        query, Wq, qp, 256, 256, 256, 0, 0, 0);

    // 2) k_proj = keys * Wk^T    (M=16384, N=256, K=256)
    //    strips = 1024 * 4 = 4096 -> 1024 blocks
    wmma_gemm_f32<true><<<dim3(1024, 1), 128, 0, stream>>>(
        keys, Wk, kp, 16384, 256, 256, 0, 0, 0);

    // 3) additive scores with native tanh (TRANS-bound stage)
    score_tanh_kernel<<<dim3(2048 / 16, 8), 256, 0, stream>>>(qp, kp, v, sc);

    // 4) softmax over Tk -> attn output
    softmax_rows_kernel<<<dim3(256), 256, 0, stream>>>(sc, attn);

    // 5) context[b] = attn[b] (32x2048) @ keys[b] (2048x256), batched over B
    //    strips per batch = 2 * 4 = 8 -> 2 blocks, grid (2, 8)
    wmma_gemm_f32<false><<<dim3(2, 8), 128, 0, stream>>>(
        attn, keys, context, 32, 256, 2048,
        (long long)32 * 2048, (long long)2048 * 256, (long long)32 * 256);
}